// Tooth_Assembler_67070209294709
// MI455X (gfx1250) — compile-verified
//
#include <hip/hip_runtime.h>

// Vector types matching the gfx1250 WMMA builtin operand shapes.
typedef float v2f __attribute__((ext_vector_type(2)));  // A/B of 16x16x4 f32 (2 VGPRs)
typedef float v8f __attribute__((ext_vector_type(8)));  // C/D 16x16 f32 (8 VGPRs)

__device__ __forceinline__ float lane_swap16(float v) {
    // swap lanes l <-> l^16 across the wave32 (executed by ALL lanes)
    return __shfl_xor(v, 16, 32);
}

// One block per (b,t). Rotate N points by R(q), recenter, translate.
// out[p] = (pred[p]-cenp[p]) @ R + ptrans + cenp[p]
__global__ __launch_bounds__(256) void tooth_assembler_wmma(
    const float* __restrict__ pred,
    const float* __restrict__ cenp,
    const float* __restrict__ dofs,
    const float* __restrict__ ptrans,
    float* __restrict__ out,
    int N)
{
    const int bt = blockIdx.x;

    // ---- quaternion -> 3x3 rotation (uniform per block; scalarized by compiler) ----
    const float qr = dofs[bt * 4 + 0];
    const float qi = dofs[bt * 4 + 1];
    const float qj = dofs[bt * 4 + 2];
    const float qk = dofs[bt * 4 + 3];
    const float two_s = 2.0f / (qr * qr + qi * qi + qj * qj + qk * qk);

    float R[3][3];
    R[0][0] = 1.0f - two_s * (qj * qj + qk * qk);
    R[0][1] = two_s * (qi * qj - qk * qr);
    R[0][2] = two_s * (qi * qk + qj * qr);
    R[1][0] = two_s * (qi * qj + qk * qr);
    R[1][1] = 1.0f - two_s * (qi * qi + qk * qk);
    R[1][2] = two_s * (qj * qk - qi * qr);
    R[2][0] = two_s * (qi * qk - qj * qr);
    R[2][1] = two_s * (qj * qk + qi * qr);
    R[2][2] = 1.0f - two_s * (qi * qi + qj * qj);

    const float tx = ptrans[bt * 3 + 0];
    const float ty = ptrans[bt * 3 + 1];
    const float tz = ptrans[bt * 3 + 2];

    const int lane = threadIdx.x & 31;
    const int wave = threadIdx.x >> 5;
    const int nwaves = blockDim.x >> 5;

    // ---- A holds R^T TWICE: rows 0..2 and rows 8..10 (16x4 f32 A layout):
    //   lane m    : v0=A(m,0) v1=A(m,1)     (K=0,1)
    //   lane m+16 : v0=A(m,2) v1=A(m,3)     (K=2,3)
    // A(j,i) = A(8+j,i) = R(i,j). Rows 8..10 make D's lanes 16-31 carry the
    // same rotated(p,j) values so the second WMMA needs no output shuffle.
    v2f A; A.x = 0.0f; A.y = 0.0f;
    {
        const int j = lane & 7;          // candidate output coordinate
        const bool arow = ((lane & 15) & 7) < 3
                          && (((lane & 15) < 3) || ((lane & 15) >= 8));
        // rows {0,1,2,8,9,10}: (lane&15) in {0,1,2,8,9,10}
        if (arow) {
            if (lane < 16) { A.x = R[0][j]; A.y = R[1][j]; }   // K=0,1
            else           { A.x = R[2][j]; }                  // K=2 (K=3 pad=0)
        }
    }

    const long base = (long)bt * (long)N;
    const int ntiles = N >> 5;  // 32 points per wave-iteration

    for (int tile = wave; tile < ntiles; tile += nwaves) {
        const long idx = base + ((long)tile << 5) + lane;  // this lane's point
        const float* pp = pred + idx * 3;
        const float* cp = cenp + idx * 3;

        // b96-coalesced loads: lane l owns 12 contiguous bytes
        const float px = pp[0], py = pp[1], pz = pp[2];
        const float cx = cp[0], cy = cp[1], cz = cp[2];
        const float x = px - cx, y = py - cy, z = pz - cz;

        // ---- B layout (4x16 f32): lanes 0-15 hold K=0,1 (v0,v1); lanes 16-31 hold K=2,3.
        // Swaps are standalone statements -> unconditional ds_bpermute + v_cndmask.
        const float zs = lane_swap16(z);
        const float xs = lane_swap16(x);
        const float ys = lane_swap16(y);
        // B0: columns = points tile*32 + 0..15
        v2f B0; B0.x = (lane < 16) ? x : zs;  B0.y = (lane < 16) ? y : 0.0f;
        // B1: columns = points tile*32 + 16..31
        v2f B1; B1.x = (lane < 16) ? xs : z;  B1.y = (lane < 16) ? ys : 0.0f;

        v8f C0 = {};
        v8f C1 = {};
        // D0: lanes 0-15,  VGPR j = rotated(point tile*32+lane, j)
        // D1: lanes 16-31, VGPR j = rotated(point tile*32+lane, j)  (via A rows 8..10)
        v8f D0 = __builtin_amdgcn_wmma_f32_16x16x4_f32(
            false, A, false, B0, (short)0, C0, false, false);
        v8f D1 = __builtin_amdgcn_wmma_f32_16x16x4_f32(
            false, A, false, B1, (short)0, C1, false, false);

        // pure v_cndmask merge -- no cross-lane ops on the output side
        const float r0 = (lane < 16) ? D0[0] : D1[0];
        const float r1 = (lane < 16) ? D0[1] : D1[1];
        const float r2 = (lane < 16) ? D0[2] : D1[2];

        // fused translate + recenter; b96-coalesced store
        float* op = out + idx * 3;
        op[0] = r0 + tx + cx;
        op[1] = r1 + ty + cy;
        op[2] = r2 + tz + cz;
    }

    // ---- scalar remainder path (N % 32 != 0); no-op for N=8192 ----
    const int rem_start = ntiles << 5;
    for (int i = rem_start + (int)threadIdx.x; i < N; i += (int)blockDim.x) {
        const long idx = base + i;
        const float px = pred[idx * 3 + 0], py = pred[idx * 3 + 1], pz = pred[idx * 3 + 2];
        const float cx = cenp[idx * 3 + 0], cy = cenp[idx * 3 + 1], cz = cenp[idx * 3 + 2];
        const float x = px - cx, y = py - cy, z = pz - cz;
        out[idx * 3 + 0] = x * R[0][0] + y * R[1][0] + z * R[2][0] + tx + cx;
        out[idx * 3 + 1] = x * R[0][1] + y * R[1][1] + z * R[2][1] + ty + cy;
        out[idx * 3 + 2] = x * R[0][2] + y * R[1][2] + z * R[2][2] + tz + cz;
    }
}

extern "C" void kernel_launch(void* const* d_in, const int* in_sizes, int n_in,
                              void* d_out, int out_size, void* d_ws, size_t ws_size,
                              hipStream_t stream) {
    (void)n_in; (void)out_size; (void)d_ws; (void)ws_size;
    const float* pred   = (const float*)d_in[0];  // [B,T,N,3]
    const float* cenp   = (const float*)d_in[1];  // [B,T,N,3]
    const float* dofs   = (const float*)d_in[2];  // [B,T,4]
    const float* ptrans = (const float*)d_in[3];  // [B,T,3]
    float* out = (float*)d_out;

    const int BT = in_sizes[2] / 4;           // B*T
    if (BT <= 0) return;
    const int N = in_sizes[0] / (BT * 3);     // points per (b,t)

    dim3 grid(BT);
    dim3 block(256);                          // 8 wave32s
    hipLaunchKernelGGL(tooth_assembler_wmma, grid, block, 0, stream,
                       pred, cenp, dofs, ptrans, out, N);
}